// GAT_63299228009396
// MI455X (gfx1250) — compile-verified
//
#include <hip/hip_runtime.h>
#include <hip/hip_bf16.h>
#include <float.h>

// GAT 3-layer forward for MI455X (gfx1250, wave32).
// Dense node-feature GEMMs use V_WMMA_F32_16X16X4_F32 (fp32 matrix pipe),
// with W staged in LDS and compile-time K/NCOLS so the K-loop is fully
// unrolled and branchless. Edge softmax/aggregation is the bandwidth-bound
// part (~5-8 GB irregular traffic @ 23.3 TB/s dominates runtime); done in 3
// passes with float atomics. Per-channel GAT bias is skipped: it cancels
// exactly under the BatchNorm mean-subtraction that follows each conv.

typedef __attribute__((ext_vector_type(2))) float v2f;
typedef __attribute__((ext_vector_type(8))) float v8f;

#define NEG_SLOPE 0.2f
#define BN_EPS 1e-5f
#define HEADS 2

// ---------------- utility ----------------

__global__ void k_fill(float* __restrict__ p, size_t cnt, float v) {
    size_t i = (size_t)blockIdx.x * blockDim.x + threadIdx.x;
    if (i < cnt) p[i] = v;
}

__device__ inline void atomicMaxF(float* addr, float val) {
    if (!(val == val)) return;  // skip NaN
    if (val >= 0.0f) atomicMax((int*)addr, __float_as_int(val));
    else             atomicMin((unsigned int*)addr, __float_as_uint(val));
}

__device__ inline void edge_sd(const int* __restrict__ ei, int E, int e, int& s, int& d) {
    if (e < E) { s = ei[e]; d = ei[E + e]; }       // edge_index[0][e], [1][e]
    else       { s = d = e - E; }                  // self loops
}

__device__ inline float lrelu(float z) { return z > 0.0f ? z : NEG_SLOPE * z; }

// ---------------- WMMA GEMM: Y[n,NCOLS] = X[n,K_DIM] @ W[K_DIM,NCOLS] --------
// Block = 256 threads = 8 waves. W tile staged in LDS (zero-padded to KP rows).
// One wave per 16x16 output tile; K-loop fully unrolled over
// V_WMMA_F32_16X16X4_F32.
//   A (16x4 f32): VGPR0 = K+0 (lanes 0-15) / K+2 (lanes 16-31); VGPR1 = K+1/K+3.
//   B (4x16 f32): mirrored row-striping across lanes (from LDS).
//   C/D (16x16 f32): VGPR r -> rows r (lanes 0-15) and r+8 (lanes 16-31).
template <int K_DIM, int NCOLS>
__global__ void k_gemm_wmma(const float* __restrict__ X, const float* __restrict__ W,
                            float* __restrict__ Y, int n, int total_tiles) {
    constexpr int KP = (K_DIM + 3) & ~3;            // K padded to multiple of 4
    constexpr int TILES_N = NCOLS / 16;
    __shared__ float wlds[KP * NCOLS];

    // cooperative stage of W into LDS (rows >= K_DIM zero-padded)
    for (int i = threadIdx.x; i < KP * NCOLS; i += blockDim.x) {
        int k = i / NCOLS;
        int c = i - k * NCOLS;
        wlds[i] = (k < K_DIM) ? W[k * NCOLS + c] : 0.0f;
    }
    __syncthreads();

    int tile = blockIdx.x * (blockDim.x >> 5) + (threadIdx.x >> 5);
    if (tile >= total_tiles) return;                 // wave-uniform
    int tm = tile / TILES_N;
    int tn = tile - tm * TILES_N;

    int lane = threadIdx.x & 31;
    int half = lane >> 4;
    int m    = lane & 15;
    int row  = tm * 16 + m;
    int col  = tn * 16 + m;
    int rload = row < n ? row : n - 1;               // branchless clamp for loads

    v8f c = {};
#pragma unroll
    for (int kk = 0; kk < KP; kk += 4) {
        int ka = kk + 2 * half;
        v2f a, b;
        if constexpr (K_DIM % 4 == 0) {
            a = *(const v2f*)(X + (size_t)rload * K_DIM + ka);   // global_load_b64
        } else {
            int i0 = ka     < K_DIM ? ka     : 0;
            int i1 = ka + 1 < K_DIM ? ka + 1 : 0;
            float ax = X[(size_t)rload * K_DIM + i0];
            float ay = X[(size_t)rload * K_DIM + i1];
            a.x = (ka     < K_DIM) ? ax : 0.0f;      // value-select, no branch
            a.y = (ka + 1 < K_DIM) ? ay : 0.0f;
        }
        b.x = wlds[ka * NCOLS + col];                // ds_load (zero-padded rows)
        b.y = wlds[(ka + 1) * NCOLS + col];
        c = __builtin_amdgcn_wmma_f32_16x16x4_f32(
                /*neg_a=*/false, a, /*neg_b=*/false, b,
                /*c_mod=*/(short)0, c, /*reuse_a=*/false, /*reuse_b=*/false);
    }

    if (tm * 16 + 16 <= n) {                         // full tile (always for N%16==0)
#pragma unroll
        for (int r = 0; r < 8; ++r)
            Y[(size_t)(tm * 16 + r + 8 * half) * NCOLS + col] = c[r];
    } else {
#pragma unroll
        for (int r = 0; r < 8; ++r) {
            int orow = tm * 16 + r + 8 * half;
            if (orow < n) Y[(size_t)orow * NCOLS + col] = c[r];
        }
    }
}

// ---------------- attention logits per node/head ----------------
__global__ void k_alphas(const float* __restrict__ h, const float* __restrict__ a_src,
                         const float* __restrict__ a_dst, float* __restrict__ alS,
                         float* __restrict__ alD, int n, int C) {
    int i = blockIdx.x * blockDim.x + threadIdx.x;
    if (i >= n * HEADS) return;
    int node = i >> 1, hh = i & 1;
    const float* hp = h + (size_t)node * HEADS * C + hh * C;
    float s = 0.0f, d = 0.0f;
    for (int c = 0; c < C; ++c) {
        float v = hp[c];
        s += v * a_src[hh * C + c];
        d += v * a_dst[hh * C + c];
    }
    alS[i] = s;
    alD[i] = d;
}

// ---------------- edge pass 1: segment max of leaky-relu scores ----------------
__global__ void k_edge_max(const float* __restrict__ alS, const float* __restrict__ alD,
                           const int* __restrict__ ei, int E, int n,
                           float* __restrict__ mmax) {
    int e = blockIdx.x * blockDim.x + threadIdx.x;
    if (e >= E + n) return;
    int s, d; edge_sd(ei, E, e, s, d);
    v2f as = *(const v2f*)(alS + 2 * (size_t)s);
    v2f ad = *(const v2f*)(alD + 2 * (size_t)d);
    atomicMaxF(&mmax[2 * d + 0], lrelu(as.x + ad.x));
    atomicMaxF(&mmax[2 * d + 1], lrelu(as.y + ad.y));
}

// ---------------- edge pass 2: exp and segment sum ----------------
__global__ void k_edge_exp(const float* __restrict__ alS, const float* __restrict__ alD,
                           const int* __restrict__ ei, int E, int n,
                           const float* __restrict__ mmax, float* __restrict__ den,
                           float* __restrict__ ex) {
    int e = blockIdx.x * blockDim.x + threadIdx.x;
    if (e >= E + n) return;
    int s, d; edge_sd(ei, E, e, s, d);
    v2f as = *(const v2f*)(alS + 2 * (size_t)s);
    v2f ad = *(const v2f*)(alD + 2 * (size_t)d);
    v2f mm = *(const v2f*)(mmax + 2 * (size_t)d);
    v2f v;
    v.x = __expf(lrelu(as.x + ad.x) - mm.x);
    v.y = __expf(lrelu(as.y + ad.y) - mm.y);
    *(v2f*)(ex + 2 * (size_t)e) = v;
    atomicAdd(&den[2 * d + 0], v.x);
    atomicAdd(&den[2 * d + 1], v.y);
}

// ---------------- edge pass 3: alpha-weighted scatter-aggregate ----------------
// One thread per (edge, 4-channel chunk); C is always a multiple of 4, so a
// chunk never crosses a head boundary and the gathered row load is one b128.
__global__ void k_edge_agg(const float* __restrict__ hsrc, const float* __restrict__ ex,
                           const float* __restrict__ den, const int* __restrict__ ei,
                           int E, int n, int C, float* __restrict__ out) {
    int dim = HEADS * C;
    int chunks = dim >> 2;
    long long idx = (long long)blockIdx.x * blockDim.x + threadIdx.x;
    long long total = (long long)(E + n) * chunks;
    if (idx >= total) return;
    int e   = (int)(idx / chunks);
    int ch4 = (int)(idx - (long long)e * chunks) * 4;
    int s, d; edge_sd(ei, E, e, s, d);
    const float4* hp = (const float4*)(hsrc + (size_t)s * dim + ch4);
    __builtin_prefetch(hp, 0, 0);                     // global_prefetch: hide gather
    int h = ch4 >= C ? 1 : 0;
    float alpha = ex[(size_t)2 * e + h] / den[2 * d + h];
    float4 hv = *hp;
    float* op = out + (size_t)d * dim + ch4;
    atomicAdd(&op[0], alpha * hv.x);
    atomicAdd(&op[1], alpha * hv.y);
    atomicAdd(&op[2], alpha * hv.z);
    atomicAdd(&op[3], alpha * hv.w);
}

// ---------------- BN stats (per channel) via LDS tree + atomics ----------------
__global__ void k_bn_stats(const float* __restrict__ x, int n, int dim,
                           float* __restrict__ sums, float* __restrict__ sqs) {
    __shared__ float ls[256];
    __shared__ float lq[256];
    int ch = blockIdx.y;
    float s = 0.0f, q = 0.0f;
    for (int i = blockIdx.x * blockDim.x + threadIdx.x; i < n; i += blockDim.x * gridDim.x) {
        float v = x[(size_t)i * dim + ch];
        s += v; q += v * v;
    }
    ls[threadIdx.x] = s; lq[threadIdx.x] = q;
    __syncthreads();
    for (int off = blockDim.x >> 1; off > 0; off >>= 1) {
        if (threadIdx.x < off) {
            ls[threadIdx.x] += ls[threadIdx.x + off];
            lq[threadIdx.x] += lq[threadIdx.x + off];
        }
        __syncthreads();
    }
    if (threadIdx.x == 0) {
        atomicAdd(&sums[ch], ls[0]);
        atomicAdd(&sqs[ch], lq[0]);
    }
}

__global__ void k_bn_apply(float* __restrict__ x, int n, int dim,
                           const float* __restrict__ sums, const float* __restrict__ sqs,
                           const float* __restrict__ g, const float* __restrict__ be) {
    size_t i = (size_t)blockIdx.x * blockDim.x + threadIdx.x;
    if (i >= (size_t)n * dim) return;
    int ch = (int)(i % dim);
    float inv_n = 1.0f / (float)n;
    float mu  = sums[ch] * inv_n;
    float var = sqs[ch] * inv_n - mu * mu;
    float v = (x[i] - mu) * rsqrtf(var + BN_EPS) * g[ch] + be[ch];
    x[i] = v > 0.0f ? v : 0.0f;
}

// ---------------- graph max-pool + FC ----------------
__global__ void k_pool(const float* __restrict__ h, const int* __restrict__ batch,
                       int n, int dim, float* __restrict__ pooled) {
    size_t i = (size_t)blockIdx.x * blockDim.x + threadIdx.x;
    if (i >= (size_t)n * dim) return;
    int node = (int)(i / dim), ch = (int)(i - (size_t)node * dim);
    atomicMaxF(&pooled[batch[node] * dim + ch], h[i]);
}

__global__ void k_fc(const float* __restrict__ pooled, const float* __restrict__ w,
                     const float* __restrict__ b, float* __restrict__ out,
                     int G, int Kdim, int O) {
    int i = blockIdx.x * blockDim.x + threadIdx.x;
    if (i >= G * O) return;
    int g = i / O, o = i - g * O;
    float acc = b[o];
    for (int k = 0; k < Kdim; ++k) acc += pooled[g * Kdim + k] * w[k * O + o];
    out[i] = acc;
}

// ---------------- host driver ----------------

static inline int divup(long long a, long long b) { return (int)((a + b - 1) / b); }

extern "C" void kernel_launch(void* const* d_in, const int* in_sizes, int n_in,
                              void* d_out, int out_size, void* d_ws, size_t ws_size,
                              hipStream_t stream) {
    const float* x   = (const float*)d_in[0];
    const int*   ei  = (const int*)d_in[1];     // [2,E] flattened
    const int*   bat = (const int*)d_in[2];
    const float* W1  = (const float*)d_in[3];
    const float* as1 = (const float*)d_in[4];
    const float* ad1 = (const float*)d_in[5];
    const float* g1  = (const float*)d_in[7];
    const float* be1 = (const float*)d_in[8];
    const float* W2  = (const float*)d_in[9];
    const float* as2 = (const float*)d_in[10];
    const float* ad2 = (const float*)d_in[11];
    const float* g2  = (const float*)d_in[13];
    const float* be2 = (const float*)d_in[14];
    const float* W3  = (const float*)d_in[15];
    const float* as3 = (const float*)d_in[16];
    const float* ad3 = (const float*)d_in[17];
    const float* g3  = (const float*)d_in[19];
    const float* be3 = (const float*)d_in[20];
    const float* fcw = (const float*)d_in[21];
    const float* fcb = (const float*)d_in[22];
    // b1/b2/b3 (d_in[6], d_in[12], d_in[18]) cancel under BN mean-subtraction.

    const int N  = in_sizes[0] / 3;
    const int E  = in_sizes[1] / 2;
    const int EE = E + N;               // edges + self loops
    const int G  = 64;
    const int MAXDIM = 128;

    // workspace carve-out
    float* P      = (float*)d_ws;                       // [N,128] gemm output h
    float* Q      = P + (size_t)N * MAXDIM;             // [N,128] agg output / layer input
    float* alS    = Q + (size_t)N * MAXDIM;             // [N,2]
    float* alD    = alS + (size_t)N * 2;
    float* mmax   = alD + (size_t)N * 2;
    float* den    = mmax + (size_t)N * 2;
    float* exbuf  = den + (size_t)N * 2;                // [EE,2]
    float* sums   = exbuf + (size_t)EE * 2;             // [128]
    float* sqs    = sums + MAXDIM;                      // [128]
    float* pooled = sqs + MAXDIM;                       // [64,128]

    auto run_layer = [&](const float* cur, int K, int C,
                         const float* W, const float* a_s, const float* a_d,
                         const float* g, const float* be) {
        const int dim = HEADS * C;
        // 1) h = cur @ W  (WMMA fp32, K/NCOLS specialized)
        int tiles = divup(N, 16) * (dim / 16);
        int gblocks = divup(tiles, 8);                  // 8 waves / block
        if (K == 3)       k_gemm_wmma<3, 32> <<<gblocks, 256, 0, stream>>>(cur, W, P, N, tiles);
        else if (K == 32) k_gemm_wmma<32, 64><<<gblocks, 256, 0, stream>>>(cur, W, P, N, tiles);
        else              k_gemm_wmma<64, 128><<<gblocks, 256, 0, stream>>>(cur, W, P, N, tiles);
        // 2) per-node attention logits
        k_alphas<<<divup((long long)N * HEADS, 256), 256, 0, stream>>>(P, a_s, a_d, alS, alD, N, C);
        // 3) init softmax state + agg output
        k_fill<<<divup((long long)N * HEADS, 256), 256, 0, stream>>>(mmax, (size_t)N * HEADS, -FLT_MAX);
        k_fill<<<divup((long long)N * HEADS, 256), 256, 0, stream>>>(den, (size_t)N * HEADS, 0.0f);
        k_fill<<<divup((long long)N * dim, 256), 256, 0, stream>>>(Q, (size_t)N * dim, 0.0f);
        // 4) segment softmax over incoming edges
        k_edge_max<<<divup(EE, 256), 256, 0, stream>>>(alS, alD, ei, E, N, mmax);
        k_edge_exp<<<divup(EE, 256), 256, 0, stream>>>(alS, alD, ei, E, N, mmax, den, exbuf);
        // 5) alpha-weighted aggregation (scatter-add)
        long long aggw = (long long)EE * (dim / 4);
        k_edge_agg<<<divup(aggw, 256), 256, 0, stream>>>(P, exbuf, den, ei, E, N, C, Q);
        // 6) BN + ReLU in place on Q
        k_fill<<<1, 256, 0, stream>>>(sums, (size_t)dim, 0.0f);
        k_fill<<<1, 256, 0, stream>>>(sqs, (size_t)dim, 0.0f);
        dim3 sgrid(32, dim);
        k_bn_stats<<<sgrid, 256, 0, stream>>>(Q, N, dim, sums, sqs);
        k_bn_apply<<<divup((long long)N * dim, 256), 256, 0, stream>>>(Q, N, dim, sums, sqs, g, be);
    };

    run_layer(x, 3,  16, W1, as1, ad1, g1, be1);   // -> Q [N,32]
    run_layer(Q, 32, 32, W2, as2, ad2, g2, be2);   // -> Q [N,64]
    run_layer(Q, 64, 64, W3, as3, ad3, g3, be3);   // -> Q [N,128]

    // graph max-pool + FC
    k_fill<<<divup((long long)G * 128, 256), 256, 0, stream>>>(pooled, (size_t)G * 128, -FLT_MAX);
    k_pool<<<divup((long long)N * 128, 256), 256, 0, stream>>>(Q, bat, N, 128, pooled);
    k_fc<<<divup((long long)G * 10, 64), 64, 0, stream>>>(pooled, fcw, fcb, (float*)d_out, G, 128, 10);
}